// LatentEwaldSum_31379031065336
// MI455X (gfx1250) — compile-verified
//
#include <hip/hip_runtime.h>
#include <math.h>

// Latent Ewald sum for MI455X (gfx1250, wave32).
// Phase matrix kr = u @ n^T done with V_WMMA_F32_16X16X4_F32 (16 atoms x 16 kvecs
// per wave-tile, K=3 padded to 4, exact f32). sincos on the TRANS pipe, segment
// sum via per-tile in-register column reduction + f32 atomics into L2-resident
// [G, 352] accumulators.

#define NK   342           // 7^3 - 1
#define NKP  352           // padded to 22 * 16
#define KT   (NKP / 16)    // 22 k-tiles of 16
#define TWOPI 6.28318530717958647692f

typedef __attribute__((ext_vector_type(2))) float v2f;
typedef __attribute__((ext_vector_type(8))) float v8f;

// Map padded k index -> integer triple, matching the reference's meshgrid('ij')
// flatten order with the (0,0,0) entry (flat index 171) removed.
__device__ __forceinline__ void k_to_n(int idx, float& nx, float& ny, float& nz) {
  if (idx >= NK) { nx = 0.f; ny = 0.f; nz = 0.f; return; }
  int m = (idx >= 171) ? idx + 1 : idx;
  int a = m / 49;
  int b = (m / 7) % 7;
  int c = m % 7;
  nx = (float)(a - 3); ny = (float)(b - 3); nz = (float)(c - 3);
}

// ---------------------------------------------------------------------------
// Kernel 1: per-graph setup. recip = 2*pi*inv(cell)^T, pref = 2*pi/|det|,
// w[g,k] = exp(-0.5*sigma^2*k2)/k2 (sigma = 1), zero sr/si accumulators.
// One block per graph.
// ---------------------------------------------------------------------------
__global__ void ewald_setup(const float* __restrict__ cell,
                            float* __restrict__ Rg, float* __restrict__ pref,
                            float* __restrict__ w,
                            float* __restrict__ sr, float* __restrict__ si) {
  const int g = blockIdx.x;
  __shared__ float Rs[9];
  if (threadIdx.x == 0) {
    const float* c = cell + g * 9;  // row-major 3x3
    float a00 = c[0], a01 = c[1], a02 = c[2];
    float a10 = c[3], a11 = c[4], a12 = c[5];
    float a20 = c[6], a21 = c[7], a22 = c[8];
    float det = a00 * (a11 * a22 - a12 * a21)
              - a01 * (a10 * a22 - a12 * a20)
              + a02 * (a10 * a21 - a11 * a20);
    float id = 1.0f / det;
    float inv[9];
    inv[0] = (a11 * a22 - a12 * a21) * id;
    inv[1] = (a02 * a21 - a01 * a22) * id;
    inv[2] = (a01 * a12 - a02 * a11) * id;
    inv[3] = (a12 * a20 - a10 * a22) * id;
    inv[4] = (a00 * a22 - a02 * a20) * id;
    inv[5] = (a02 * a10 - a00 * a12) * id;
    inv[6] = (a10 * a21 - a11 * a20) * id;
    inv[7] = (a01 * a20 - a00 * a21) * id;
    inv[8] = (a00 * a11 - a01 * a10) * id;
    // recip[r][c] = 2*pi * inv[c][r]
    for (int r = 0; r < 3; ++r)
      for (int cc = 0; cc < 3; ++cc) {
        float v = TWOPI * inv[cc * 3 + r];
        Rs[r * 3 + cc] = v;
        Rg[g * 9 + r * 3 + cc] = v;
      }
    pref[g] = TWOPI / fabsf(det);
  }
  __syncthreads();
  for (int k = threadIdx.x; k < NKP; k += blockDim.x) {
    float nx, ny, nz;
    k_to_n(k, nx, ny, nz);
    // kvec_d = sum_j n_j * R[j][d]
    float kv0 = nx * Rs[0] + ny * Rs[3] + nz * Rs[6];
    float kv1 = nx * Rs[1] + ny * Rs[4] + nz * Rs[7];
    float kv2 = nx * Rs[2] + ny * Rs[5] + nz * Rs[8];
    float k2 = kv0 * kv0 + kv1 * kv1 + kv2 * kv2;
    w[g * NKP + k] = (k < NK) ? __expf(-0.5f * k2) / k2 : 0.0f;
    sr[g * NKP + k] = 0.0f;
    si[g * NKP + k] = 0.0f;
  }
}

// ---------------------------------------------------------------------------
// Kernel 2: structure factor accumulation. Each wave owns 16 atoms; loops over
// 22 k-tiles of 16 kvecs. kr tile via WMMA f32 16x16x4 (A = u per-atom rows,
// B = integer n-grid columns), sincos per element, column-sum over the atom
// dim in-register, atomicAdd into sr/si[g, k].
// ---------------------------------------------------------------------------
__global__ void ewald_sk(const float* __restrict__ q, const float* __restrict__ pos,
                         const int* __restrict__ batch, const float* __restrict__ Rg,
                         float* __restrict__ sr, float* __restrict__ si, int N) {
  const int lane = threadIdx.x & 31;
  const int wv = blockIdx.x * (blockDim.x >> 5) + (threadIdx.x >> 5);
  const int abase = wv * 16;
  if (abase >= N) return;                 // wave-uniform exit

  const int am = abase + (lane & 15);     // both half-waves mirror the 16 atoms
  const bool valid = am < N;
  const int ai = valid ? am : (N - 1);
  const float qa = valid ? q[ai] : 0.0f;  // pad atoms contribute 0
  const int ga = batch[ai];
  const float px = pos[ai * 3 + 0];
  const float py = pos[ai * 3 + 1];
  const float pz = pos[ai * 3 + 2];

  const float* R = Rg + ga * 9;           // L2-hot, near-uniform per wave
  const float ux = R[0] * px + R[1] * py + R[2] * pz;
  const float uy = R[3] * px + R[4] * py + R[5] * pz;
  const float uz = R[6] * px + R[7] * py + R[8] * pz;

  // A fragment (16x4 f32): lanes 0-15 -> (K0,K1)=(ux,uy); lanes 16-31 -> (K2,K3)=(uz,0)
  v2f A;
  A.x = (lane < 16) ? ux : uz;
  A.y = (lane < 16) ? uy : 0.0f;

  // batch is sorted: graphs present in this tile span [g0, g1] (almost always g0==g1)
  const int g0 = __shfl(ga, 0, 32);
  const int g1 = __shfl(ga, 15, 32);
  const int moff = (lane < 16) ? 0 : 8;   // D rows M held by this lane: moff..moff+7

  for (int gg = g0; gg <= g1; ++gg) {
    const float qg = (ga == gg) ? qa : 0.0f;   // mask atoms outside graph gg
    float qm[8];
#pragma unroll
    for (int v = 0; v < 8; ++v) qm[v] = __shfl(qg, v + moff, 32);

    float* __restrict__ srg = sr + gg * NKP;
    float* __restrict__ sig = si + gg * NKP;

    for (int t = 0; t < KT; ++t) {
      // B fragment (4x16 f32): column j = t*16 + (lane&15); integer n exact in f32
      float nx, ny, nz;
      k_to_n(t * 16 + (lane & 15), nx, ny, nz);
      v2f B;
      B.x = (lane < 16) ? nx : nz;
      B.y = (lane < 16) ? ny : 0.0f;

      v8f D = {0.f, 0.f, 0.f, 0.f, 0.f, 0.f, 0.f, 0.f};
      // D[M][N] = kr for (atom abase+M, kvec t*16+N)
      D = __builtin_amdgcn_wmma_f32_16x16x4_f32(
          /*neg_a=*/false, A, /*neg_b=*/false, B,
          /*c_mod=*/(short)0, D, /*reuse_a=*/false, /*reuse_b=*/false);

      float csum = 0.0f, ssum = 0.0f;
#pragma unroll
      for (int v = 0; v < 8; ++v) {
        float kr = D[v];
        csum += qm[v] * __cosf(kr);
        ssum += qm[v] * __sinf(kr);
      }
      // fold the other 8 atoms (rows M^8) held by the partner half-wave
      csum += __shfl_xor(csum, 16, 32);
      ssum += __shfl_xor(ssum, 16, 32);
      if (lane < 16) {
        atomicAdd(&srg[t * 16 + lane], csum);
        atomicAdd(&sig[t * 16 + lane], ssum);
      }
    }
  }
}

// ---------------------------------------------------------------------------
// Kernel 3: e[g] = pref[g] * sum_k w[g,k] * (sr^2 + si^2). One block per graph.
// ---------------------------------------------------------------------------
__global__ void ewald_finalize(const float* __restrict__ w,
                               const float* __restrict__ sr,
                               const float* __restrict__ si,
                               const float* __restrict__ pref,
                               float* __restrict__ out) {
  const int g = blockIdx.x;
  float acc = 0.0f;
  for (int k = threadIdx.x; k < NKP; k += blockDim.x) {
    float a = sr[g * NKP + k];
    float b = si[g * NKP + k];
    acc += w[g * NKP + k] * (a * a + b * b);
  }
  __shared__ float red[128];
  red[threadIdx.x] = acc;
  __syncthreads();
  for (int s = 64; s > 0; s >>= 1) {
    if (threadIdx.x < s) red[threadIdx.x] += red[threadIdx.x + s];
    __syncthreads();
  }
  if (threadIdx.x == 0) out[g] = pref[g] * red[0];
}

extern "C" void kernel_launch(void* const* d_in, const int* in_sizes, int n_in,
                              void* d_out, int out_size, void* d_ws, size_t ws_size,
                              hipStream_t stream) {
  const float* q    = (const float*)d_in[0];  // [N,1]
  const float* pos  = (const float*)d_in[1];  // [N,3]
  const int*   bat  = (const int*)d_in[2];    // [N]
  const float* cell = (const float*)d_in[3];  // [G,3,3]
  float* out = (float*)d_out;                 // [G]

  const int N = in_sizes[0];
  const int G = in_sizes[3] / 9;

  // Workspace layout (floats): Rg[G*9] | pref[G] | w[G*NKP] | sr[G*NKP] | si[G*NKP]
  float* ws   = (float*)d_ws;
  float* Rg   = ws;
  float* pref = Rg + (size_t)G * 9;
  float* w    = pref + G;
  float* sr   = w  + (size_t)G * NKP;
  float* si   = sr + (size_t)G * NKP;

  hipLaunchKernelGGL(ewald_setup, dim3(G), dim3(128), 0, stream,
                     cell, Rg, pref, w, sr, si);

  const int waves  = (N + 15) / 16;
  const int blocks = (waves + 7) / 8;       // 8 waves (256 threads) per block
  hipLaunchKernelGGL(ewald_sk, dim3(blocks), dim3(256), 0, stream,
                     q, pos, bat, Rg, sr, si, N);

  hipLaunchKernelGGL(ewald_finalize, dim3(G), dim3(128), 0, stream,
                     w, sr, si, pref, out);
}